// GCNGenerator_9191230014151
// MI455X (gfx1250) — compile-verified
//
#include <hip/hip_runtime.h>
#include <math.h>

typedef float v2f __attribute__((ext_vector_type(2)));
typedef float v8f __attribute__((ext_vector_type(8)));

#define BN_EPS 1e-3f

// ---- CDNA5 async global->LDS copy (ASYNCcnt-tracked), via inline asm.
// GLOBAL_LOAD_ASYNC_TO_LDS_B128, GV mode: LDS destination address is taken from the VDST VGPR;
// generic pointers to __shared__ carry the LDS offset in their low 32 bits.
__device__ __forceinline__ void async_b128(const void* g, void* l) {
  unsigned lds_off = (unsigned)(size_t)l;
  asm volatile("global_load_async_to_lds_b128 %0, %1, off"
               :: "v"(lds_off), "v"(g)
               : "memory");
}
__device__ __forceinline__ void wait_async0() {
  asm volatile("s_wait_asynccnt 0x0" ::: "memory");
}

// ---------------------------------------------------------------- utilities
__global__ void fill_kernel(float* p, float v, long long n) {
  long long i = (long long)blockIdx.x * blockDim.x + threadIdx.x;
  if (i < n) p[i] = v;
}

__global__ void deg_accum_kernel(float* deg, const int* col, const float* w, int E) {
  int e = blockIdx.x * blockDim.x + threadIdx.x;
  if (e < E) atomicAdd(&deg[col[e]], w[e]);
}

__global__ void dinv_kernel(const float* deg, float* dinv, float* selfn, int n) {
  int i = blockIdx.x * blockDim.x + threadIdx.x;
  if (i < n) {
    float d = deg[i];
    float r = d > 0.0f ? rsqrtf(d) : 0.0f;
    dinv[i]  = r;
    selfn[i] = r * r;   // self-loop norm: dinv[i] * 1 * dinv[i]
  }
}

__global__ void edge_norm_kernel(const int* row, const int* col, const float* w,
                                 const float* dinv, float* nrm, int E) {
  int e = blockIdx.x * blockDim.x + threadIdx.x;
  if (e < E) nrm[e] = dinv[row[e]] * w[e] * dinv[col[e]];
}

// ------------------------------------------------- dense GEMM: hw = A[M,K] @ B[K,D]
// block tile 128(M) x 64(N), 256 threads = 8 waves; wave w -> rows [w*16, w*16+16),
// 4 accumulators of 16x16 across the 64 N cols. fp32 WMMA 16x16x4.
// Epilogue also writes Out = bias + selfnorm * hw (self-loop + bias, fused).
#define TM 128
#define TN 64
#define TK 16
#define APAD 4   // A LDS stride 20 floats: 16B-aligned rows, conflict-free frag reads
#define BPAD 8   // B LDS stride 72 floats: 16B-aligned, half-wave bank-disjoint
__global__ __launch_bounds__(256)
void gemm_wmma_kernel(const float* __restrict__ A, const float* __restrict__ B,
                      float* __restrict__ C, float* __restrict__ Out,
                      const float* __restrict__ bias, const float* __restrict__ selfn,
                      int M, int K, int D) {
  __shared__ float As[TM][TK + APAD];
  __shared__ float Bs[TK][TN + BPAD];

  const int tid  = threadIdx.x;
  const int wave = tid >> 5;
  const int lane = tid & 31;
  const int m0 = blockIdx.x * TM;
  const int n0 = blockIdx.y * TN;

  v8f acc[4] = {};

  for (int k0 = 0; k0 < K; k0 += TK) {
    // async global->LDS staging: A tile 128x16 (2 x b128/thread), B tile 16x64 (1/thread)
#pragma unroll
    for (int i = 0; i < 2; ++i) {
      int c  = tid + i * 256;
      int r  = c >> 2;
      int cg = (c & 3) * 4;
      int gr = m0 + r;
      if (gr >= M) gr = M - 1;   // clamp: junk rows never stored
      async_b128(&A[(size_t)gr * K + k0 + cg], &As[r][cg]);
    }
    {
      int r  = tid >> 4;
      int cg = (tid & 15) * 4;
      async_b128(&B[(size_t)(k0 + r) * D + n0 + cg], &Bs[r][cg]);
    }
    wait_async0();
    __syncthreads();

    const int mrow = wave * 16 + (lane & 15);
    const int koff = (lane >> 4) * 2;     // ISA 16x4 A layout: lanes16-31 hold K+2
    const int ncol = lane & 15;
#pragma unroll
    for (int kk = 0; kk < 4; ++kk) {
      const int kb = kk * 4 + koff;
      v2f a;
      a.x = As[mrow][kb];
      a.y = As[mrow][kb + 1];
#pragma unroll
      for (int j = 0; j < 4; ++j) {
        v2f b;
        b.x = Bs[kb][j * 16 + ncol];
        b.y = Bs[kb + 1][j * 16 + ncol];
        acc[j] = __builtin_amdgcn_wmma_f32_16x16x4_f32(
            false, a, false, b, (short)0, acc[j], false, false);
      }
    }
    __syncthreads();
  }

  // epilogue: C/D layout -> VGPR r: M = r (lanes 0-15), M = 8+r (lanes 16-31)
  const int mbase = m0 + wave * 16 + ((lane >> 4) * 8);
  const int ncol  = lane & 15;
  float sn[8];
#pragma unroll
  for (int r = 0; r < 8; ++r)
    sn[r] = (mbase + r < M) ? selfn[mbase + r] : 0.0f;
#pragma unroll
  for (int j = 0; j < 4; ++j) {
    const float bj = bias[n0 + j * 16 + ncol];
#pragma unroll
    for (int r = 0; r < 8; ++r) {
      int gr = mbase + r;
      if (gr < M) {
        float val = acc[j][r];
        size_t idx = (size_t)gr * D + n0 + j * 16 + ncol;
        C[idx]   = val;                 // raw hw (edge gather source)
        Out[idx] = bj + sn[r] * val;    // bias + self-loop contribution
      }
    }
  }
}

// out[col] += norm[e] * hw[row]; edge-major feature-minor for coalesced gather/atomics
__global__ void edge_scatter_kernel(const float* __restrict__ hw, float* __restrict__ out,
                                    const int* __restrict__ row, const int* __restrict__ col,
                                    const float* __restrict__ nrm,
                                    long long total, int ld, int fmask, int D) {
  long long i = (long long)blockIdx.x * blockDim.x + threadIdx.x;
  if (i >= total) return;
  int f = (int)(i & fmask);
  int e = (int)(i >> ld);
  float v = nrm[e] * hw[(size_t)row[e] * D + f];
  atomicAdd(&out[(size_t)col[e] * D + f], v);
}

__global__ void bn_sigmoid_kernel(float* __restrict__ h,
                                  const float* __restrict__ g, const float* __restrict__ be,
                                  const float* __restrict__ m, const float* __restrict__ v,
                                  long long total, int fmask, int has_bn) {
  long long i = (long long)blockIdx.x * blockDim.x + threadIdx.x;
  if (i >= total) return;
  int f = (int)(i & fmask);
  float t = h[i];
  if (has_bn) t = (t - m[f]) * rsqrtf(v[f] + BN_EPS) * g[f] + be[f];
  h[i] = 1.0f / (1.0f + expf(-t));
}

// ------------------------------------------------ G = h.T @ h  (split-K WMMA + atomics)
#define HTCHUNK 320   // 20 k-steps of 16 per block
#define HPAD 8        // hs stride 136 floats: 16B-aligned, bank-disjoint half-waves
__global__ __launch_bounds__(256)
void htg_wmma_kernel(const float* __restrict__ h, float* __restrict__ G, int N) {
  __shared__ float hs[16][128 + HPAD];
  const int tid  = threadIdx.x;
  const int wave = tid >> 5;
  const int lane = tid & 31;
  const int kbeg = blockIdx.x * HTCHUNK;
  const int kend = min(kbeg + HTCHUNK, N);

  v8f acc[8] = {};

  for (int k0 = kbeg; k0 < kend; k0 += 16) {
    if (k0 + 16 <= N) {   // full tile: async copy straight into LDS
#pragma unroll
      for (int i = 0; i < 2; ++i) {
        int c  = tid + i * 256;
        int r  = c >> 5;
        int cg = (c & 31) * 4;
        async_b128(&h[(size_t)(k0 + r) * 128 + cg], &hs[r][cg]);
      }
      wait_async0();
    } else {              // K tail: zero-padded sync path (atomics must see zeros)
#pragma unroll
      for (int i = 0; i < 2; ++i) {
        int c  = tid + i * 256;
        int r  = c >> 5;
        int cg = (c & 31) * 4;
        int gr = k0 + r;
        float4 v = make_float4(0.f, 0.f, 0.f, 0.f);
        if (gr < N) v = *(const float4*)&h[(size_t)gr * 128 + cg];
        hs[r][cg + 0] = v.x; hs[r][cg + 1] = v.y;
        hs[r][cg + 2] = v.z; hs[r][cg + 3] = v.w;
      }
    }
    __syncthreads();

    const int m0   = wave * 16;
    const int koff = (lane >> 4) * 2;
    const int c15  = lane & 15;
#pragma unroll
    for (int kk = 0; kk < 4; ++kk) {
      const int kb = kk * 4 + koff;
      v2f a;                       // A = h.T : A[m][k] = h[k][m]
      a.x = hs[kb][m0 + c15];
      a.y = hs[kb + 1][m0 + c15];
#pragma unroll
      for (int j = 0; j < 8; ++j) {
        v2f b;                     // B = h : B[k][n] = h[k][n]
        b.x = hs[kb][j * 16 + c15];
        b.y = hs[kb + 1][j * 16 + c15];
        acc[j] = __builtin_amdgcn_wmma_f32_16x16x4_f32(
            false, a, false, b, (short)0, acc[j], false, false);
      }
    }
    __syncthreads();
  }

  const int mbase = wave * 16 + ((lane >> 4) * 8);
  const int c15   = lane & 15;
#pragma unroll
  for (int j = 0; j < 8; ++j)
#pragma unroll
    for (int r = 0; r < 8; ++r)
      atomicAdd(&G[(size_t)(mbase + r) * 128 + j * 16 + c15], acc[j][r]);
}

__global__ void finalize_kernel(const float* __restrict__ G, float* __restrict__ out) {
  int idx = blockIdx.x * blockDim.x + threadIdx.x;  // 16384 threads
  if (idx >= 128 * 128) return;
  int i = idx >> 7, j = idx & 127;
  out[idx] = (i == j) ? 0.0f : 0.5f * (G[i * 128 + j] + G[j * 128 + i]);
}

// ---------------------------------------------------------------- host driver
extern "C" void kernel_launch(void* const* d_in, const int* in_sizes, int n_in,
                              void* d_out, int out_size, void* d_ws, size_t ws_size,
                              hipStream_t stream) {
  (void)n_in; (void)out_size; (void)ws_size;
  const float* x  = (const float*)d_in[0];
  const int*   ei = (const int*)d_in[1];
  const float* ea = (const float*)d_in[2];
  const float *W[5], *b[5];
  for (int i = 0; i < 5; ++i) { W[i] = (const float*)d_in[3 + 2 * i]; b[i] = (const float*)d_in[4 + 2 * i]; }
  const float *g[4], *be[4], *m[4], *v[4];
  for (int i = 0; i < 4; ++i) {
    g[i]  = (const float*)d_in[13 + 4 * i];
    be[i] = (const float*)d_in[14 + 4 * i];
    m[i]  = (const float*)d_in[15 + 4 * i];
    v[i]  = (const float*)d_in[16 + 4 * i];
  }

  const int N = in_sizes[0] / 128;
  const int E = in_sizes[2];
  const int* row  = ei;       // edge_index[0]
  const int* colv = ei + E;   // edge_index[1]

  float* ws    = (float*)d_ws;
  float* deg   = ws;  ws += N;
  float* dinv  = ws;  ws += N;
  float* selfn = ws;  ws += N;
  float* nrm   = ws;  ws += E;
  float* G     = ws;  ws += 128 * 128;
  float* hw    = ws;  ws += (size_t)N * 512;
  float* hA    = ws;  ws += (size_t)N * 512;
  float* hB    = ws;  ws += (size_t)N * 512;

  // gcn_norm (cached, shared by all 5 layers)
  fill_kernel<<<(N + 255) / 256, 256, 0, stream>>>(deg, 1.0f, N);     // self-loop weight
  deg_accum_kernel<<<(E + 255) / 256, 256, 0, stream>>>(deg, colv, ea, E);
  dinv_kernel<<<(N + 255) / 256, 256, 0, stream>>>(deg, dinv, selfn, N);
  edge_norm_kernel<<<(E + 255) / 256, 256, 0, stream>>>(row, colv, ea, dinv, nrm, E);

  const int dims[6] = {128, 256, 512, 256, 128, 128};
  const float* hin = x;
  float* hbuf[2] = {hA, hB};
  for (int l = 0; l < 5; ++l) {
    const int K = dims[l], D = dims[l + 1];
    float* hout = hbuf[l & 1];
    const int ld = 31 - __builtin_clz((unsigned)D);

    dim3 gg((N + TM - 1) / TM, D / TN);
    gemm_wmma_kernel<<<gg, 256, 0, stream>>>(hin, W[l], hw, hout, b[l], selfn, N, K, D);

    long long etot = (long long)E * D;
    edge_scatter_kernel<<<(unsigned)((etot + 255) / 256), 256, 0, stream>>>(
        hw, hout, row, colv, nrm, etot, ld, D - 1, D);

    long long tot = (long long)N * D;
    bn_sigmoid_kernel<<<(unsigned)((tot + 255) / 256), 256, 0, stream>>>(
        hout, (l < 4) ? g[l] : nullptr, (l < 4) ? be[l] : nullptr,
        (l < 4) ? m[l] : nullptr, (l < 4) ? v[l] : nullptr,
        tot, D - 1, (l < 4) ? 1 : 0);

    hin = hout;
  }

  // Gram matrix + symmetrize / zero diagonal
  fill_kernel<<<(16384 + 255) / 256, 256, 0, stream>>>(G, 0.0f, 16384);
  htg_wmma_kernel<<<(N + HTCHUNK - 1) / HTCHUNK, 256, 0, stream>>>(hin, G, N);
  finalize_kernel<<<64, 256, 0, stream>>>(G, (float*)d_out);
}